// MMModel_23854248362073
// MI455X (gfx1250) — compile-verified
//
#include <hip/hip_runtime.h>
#include <hip/hip_bf16.h>

// ---------------------------------------------------------------------------
// Model dims (fixed by the reference)
// ---------------------------------------------------------------------------
#define Bsz 4
#define Lsz 64
#define DMsz 1024
#define Hsz 16
#define DKsz 64
#define DVsz 64
#define Dsz 1024           // H*DV
#define DIsz 4096
#define Rsz 256            // B*L rows
#define NEGC (-1.0e9f)

typedef float v2f __attribute__((ext_vector_type(2)));
typedef float v8f __attribute__((ext_vector_type(8)));

// ---------------------------------------------------------------------------
// fp32 WMMA GEMM:  C[M,N] = A[M,K] @ B[K,N]  (row-major, lda=K, ldb=ldc=N),
// optional bias / relu / residual epilogue.
//
// Block: 256 threads = 8 waves laid out 2(M) x 4(N).
// Block tile: 64(M) x 128(N), K staged 64 at a time through LDS.
// Wave tile: 32x32 = 2x2 subtiles of 16x16 -> 4 v8f accumulators, so each
// A/B fragment is reused twice (1.0 ds_load_2addr per v_wmma).
// Fragments for 8 K-steps are staged into distinct registers before a burst
// of 32 back-to-back V_WMMA_F32_16X16X4_F32; next K-chunk's global loads are
// prefetched into registers during the compute phase.
// ---------------------------------------------------------------------------
#define GEMM_BIAS  1
#define GEMM_RELU  2
#define GEMM_RESID 4

__global__ __launch_bounds__(256)
void gemm_wmma_f32(const float* __restrict__ A, const float* __restrict__ B,
                   const float* __restrict__ bias, const float* __restrict__ resid,
                   float* __restrict__ C, int M, int N, int K, int flags)
{
    __shared__ float As[64][64 + 1];     // +1 pad: conflict-free fragment reads
    __shared__ float Bs[64][128 + 1];

    const int tid   = threadIdx.x;
    const int wave  = tid >> 5;
    const int lane  = tid & 31;
    const int l15   = lane & 15;
    const int lhalf = lane >> 4;                 // 0: lanes 0-15, 1: lanes 16-31
    const int m0    = blockIdx.y * 64;
    const int nblk  = blockIdx.x * 128;
    const int mq    = wave >> 2;                 // 0..1  -> 32-row half
    const int nq    = wave & 3;                  // 0..3  -> 32-col slice

    // global-load register staging (double buffer against LDS)
    const int ar = tid >> 4;                     // A: rows ar, ar+16, ar+32, ar+48
    const int ac = (tid & 15) << 2;              // A: col 0..60
    const int br = (tid >> 5) * 8;               // B: rows br..br+7
    const int bc = (tid & 31) << 2;              // B: col 0..124
    float4 aprf[4], bprf[8];

    v8f acc[2][2];
#pragma unroll
    for (int x = 0; x < 2; ++x)
#pragma unroll
        for (int y = 0; y < 2; ++y)
            acc[x][y] = v8f{0.f, 0.f, 0.f, 0.f, 0.f, 0.f, 0.f, 0.f};

    // prologue: fetch K-chunk 0
#pragma unroll
    for (int rr = 0; rr < 4; ++rr)
        aprf[rr] = *(const float4*)(A + (size_t)(m0 + ar + rr * 16) * K + ac);
#pragma unroll
    for (int rr = 0; rr < 8; ++rr)
        bprf[rr] = *(const float4*)(B + (size_t)(br + rr) * N + nblk + bc);

    const int arow0 = mq * 32 + l15;             // + msub*16
    const int bcol0 = nq * 32 + l15;             // + nsub*16

    for (int k0 = 0; k0 < K; k0 += 64) {
        // --- commit staged registers to LDS
#pragma unroll
        for (int rr = 0; rr < 4; ++rr) {
            As[ar + rr * 16][ac + 0] = aprf[rr].x;
            As[ar + rr * 16][ac + 1] = aprf[rr].y;
            As[ar + rr * 16][ac + 2] = aprf[rr].z;
            As[ar + rr * 16][ac + 3] = aprf[rr].w;
        }
#pragma unroll
        for (int rr = 0; rr < 8; ++rr) {
            Bs[br + rr][bc + 0] = bprf[rr].x;
            Bs[br + rr][bc + 1] = bprf[rr].y;
            Bs[br + rr][bc + 2] = bprf[rr].z;
            Bs[br + rr][bc + 3] = bprf[rr].w;
        }
        __syncthreads();

        // --- prefetch next K-chunk (overlaps with WMMA phase below)
        if (k0 + 64 < K) {
#pragma unroll
            for (int rr = 0; rr < 4; ++rr)
                aprf[rr] = *(const float4*)(A + (size_t)(m0 + ar + rr * 16) * K
                                            + k0 + 64 + ac);
#pragma unroll
            for (int rr = 0; rr < 8; ++rr)
                bprf[rr] = *(const float4*)(B + (size_t)(k0 + 64 + br + rr) * N
                                            + nblk + bc);
        }

        // --- compute: two half-chunks of 8 K-steps, fragments fully staged
#pragma unroll
        for (int half = 0; half < 2; ++half) {
            v2f af[2][8], bf[2][8];
#pragma unroll
            for (int s = 0; s < 8; ++s) {
                const int kb = half * 32 + s * 4 + lhalf * 2;   // ISA fp32 frag layout
                af[0][s] = v2f{As[arow0][kb],      As[arow0][kb + 1]};
                af[1][s] = v2f{As[arow0 + 16][kb], As[arow0 + 16][kb + 1]};
                bf[0][s] = v2f{Bs[kb][bcol0],      Bs[kb + 1][bcol0]};
                bf[1][s] = v2f{Bs[kb][bcol0 + 16], Bs[kb + 1][bcol0 + 16]};
            }
#pragma unroll
            for (int s = 0; s < 8; ++s) {
                acc[0][0] = __builtin_amdgcn_wmma_f32_16x16x4_f32(
                    false, af[0][s], false, bf[0][s], (short)0, acc[0][0], false, false);
                acc[0][1] = __builtin_amdgcn_wmma_f32_16x16x4_f32(
                    false, af[0][s], false, bf[1][s], (short)0, acc[0][1], false, false);
                acc[1][0] = __builtin_amdgcn_wmma_f32_16x16x4_f32(
                    false, af[1][s], false, bf[0][s], (short)0, acc[1][0], false, false);
                acc[1][1] = __builtin_amdgcn_wmma_f32_16x16x4_f32(
                    false, af[1][s], false, bf[1][s], (short)0, acc[1][1], false, false);
            }
        }
        __syncthreads();
    }

    // --- epilogue + store
#pragma unroll
    for (int nsub = 0; nsub < 2; ++nsub) {
        const int col = nblk + nq * 32 + nsub * 16 + l15;
        const float bv = (flags & GEMM_BIAS) ? bias[col] : 0.f;
#pragma unroll
        for (int msub = 0; msub < 2; ++msub) {
#pragma unroll
            for (int r = 0; r < 8; ++r) {
                const int row = m0 + mq * 32 + msub * 16 + r + lhalf * 8;
                float v = acc[msub][nsub][r] + bv;
                if (flags & GEMM_RESID) v += resid[(size_t)row * N + col];
                if (flags & GEMM_RELU)  v = fmaxf(v, 0.f);
                C[(size_t)row * N + col] = v;
            }
        }
    }
}

// ---------------------------------------------------------------------------
// Decision gate: dec[i][j][b][m][n] = NEG * (s > 0),
// s = sum_c relu(a1_j[b,m,c] + a0_i[b,n,c] + b1[c]) * (w2[c,1]-w2[c,0]) + (b2[1]-b2[0])
// One wave32 per output element; lane-strided coalesced reads; shfl reduce.
// ---------------------------------------------------------------------------
__global__ __launch_bounds__(256)
void dec_kernel(const float* __restrict__ a0base, const float* __restrict__ a1base,
                const float* __restrict__ b1, const float* __restrict__ w2,
                const float* __restrict__ b2, float* __restrict__ dec)
{
    const int wid  = (int)((blockIdx.x * 256u + threadIdx.x) >> 5);
    const int lane = threadIdx.x & 31;
    const int n    = wid & 63;
    const int m    = (wid >> 6) & 63;
    const int b    = (wid >> 12) & 3;
    const int pair = wid >> 14;                  // 0..8  (= i*3 + j)
    const int i    = pair / 3;
    const int j    = pair % 3;

    const float* a1 = a1base + ((size_t)j * Rsz + b * Lsz + m) * 2048;
    const float* a0 = a0base + ((size_t)i * Rsz + b * Lsz + n) * 2048;

    float acc = 0.f;
    for (int c = lane; c < 2048; c += 32) {
        float t = a1[c] + a0[c] + b1[c];
        if (t > 0.f) acc += t * (w2[2 * c + 1] - w2[2 * c]);
    }
#pragma unroll
    for (int off = 16; off; off >>= 1) acc += __shfl_xor(acc, off, 32);
    if (lane == 0) {
        float s = acc + (b2[1] - b2[0]);
        dec[wid] = (s > 0.f) ? NEGC : 0.f;
    }
}

// ---------------------------------------------------------------------------
// Attention scores + softmax over 192 keys. One block (192 threads) per
// (i, b, h, q). S[i][b][h][q][j*64+k] = softmax(q.k/8 + dec + mask)
// ---------------------------------------------------------------------------
__global__ __launch_bounds__(192)
void attn_scores_kernel(const float* __restrict__ Qb, const float* __restrict__ Kb,
                        const float* __restrict__ dec, const int* __restrict__ mask,
                        float* __restrict__ S)
{
    const int blk = blockIdx.x;
    const int q = blk & 63;
    const int h = (blk >> 6) & 15;
    const int b = (blk >> 10) & 3;
    const int i = blk >> 12;
    const int t = threadIdx.x;                   // 0..191
    const int j = t >> 6;
    const int k = t & 63;

    __shared__ float qrow[64];
    __shared__ float red[192];

    const float* Q = Qb + ((size_t)i * Rsz + b * Lsz + q) * DMsz + h * DKsz;
    if (t < 64) qrow[t] = Q[t];
    __syncthreads();

    const float* Kp = Kb + ((size_t)j * Rsz + b * Lsz + k) * DMsz + h * DKsz;
    float s = 0.f;
#pragma unroll 8
    for (int d = 0; d < 64; ++d) s += qrow[d] * Kp[d];
    s *= 0.125f;                                  // 1/sqrt(DK)
    s += dec[(((size_t)(i * 3 + j) * Bsz + b) * Lsz + q) * Lsz + k];
    if (mask[((size_t)b * Lsz + q) * Lsz + k] == 0) s = NEGC;

    // softmax over 192
    red[t] = s; __syncthreads();
    if (t < 64) red[t] = fmaxf(red[t], fmaxf(red[t + 64], red[t + 128]));
    __syncthreads();
    for (int st = 32; st > 0; st >>= 1) {
        if (t < st) red[t] = fmaxf(red[t], red[t + st]);
        __syncthreads();
    }
    const float mx = red[0];
    __syncthreads();
    const float e = __expf(s - mx);
    red[t] = e; __syncthreads();
    if (t < 64) red[t] = red[t] + red[t + 64] + red[t + 128];
    __syncthreads();
    for (int st = 32; st > 0; st >>= 1) {
        if (t < st) red[t] += red[t + st];
        __syncthreads();
    }
    S[(size_t)blk * 192 + t] = e * (1.0f / red[0]);
}

// ---------------------------------------------------------------------------
// o = A @ V : one block (64 threads = lanes over d) per (i,b,h,q).
// Coalesced 256B V row reads per key. Writes O in (R,1024) layout.
// ---------------------------------------------------------------------------
__global__ __launch_bounds__(64)
void attn_av_kernel(const float* __restrict__ S, const float* __restrict__ Vb,
                    float* __restrict__ Ob)
{
    const int blk = blockIdx.x;
    const int q = blk & 63;
    const int h = (blk >> 6) & 15;
    const int b = (blk >> 10) & 3;
    const int i = blk >> 12;
    const int d = threadIdx.x;

    __shared__ float arow[192];
    const float* Sp = S + (size_t)blk * 192;
    arow[d] = Sp[d]; arow[d + 64] = Sp[d + 64]; arow[d + 128] = Sp[d + 128];
    __syncthreads();

    float acc = 0.f;
    for (int jk = 0; jk < 192; ++jk) {
        int j = jk >> 6, k = jk & 63;
        acc += arow[jk] * Vb[((size_t)j * Rsz + b * Lsz + k) * DMsz + h * DVsz + d];
    }
    Ob[((size_t)i * Rsz + b * Lsz + q) * DMsz + h * DVsz + d] = acc;
}

// ---------------------------------------------------------------------------
// LayerNorm over 1024-wide rows, optional residual add first.
// One 256-thread block per row.
// ---------------------------------------------------------------------------
__global__ __launch_bounds__(256)
void layernorm_kernel(const float* __restrict__ x, const float* __restrict__ resid,
                      const float* __restrict__ g, const float* __restrict__ bta,
                      float* __restrict__ out)
{
    const int row = blockIdx.x;
    const int t = threadIdx.x;
    __shared__ float red[256];

    const float* xp = x + (size_t)row * 1024;
    float v[4];
    float s = 0.f;
#pragma unroll
    for (int u = 0; u < 4; ++u) {
        float val = xp[t + u * 256];
        if (resid) val += resid[(size_t)row * 1024 + t + u * 256];
        v[u] = val; s += val;
    }
    red[t] = s; __syncthreads();
    for (int st = 128; st > 0; st >>= 1) { if (t < st) red[t] += red[t + st]; __syncthreads(); }
    const float mu = red[0] * (1.f / 1024.f);
    __syncthreads();

    float s2 = 0.f;
#pragma unroll
    for (int u = 0; u < 4; ++u) { float dd = v[u] - mu; s2 += dd * dd; }
    red[t] = s2; __syncthreads();
    for (int st = 128; st > 0; st >>= 1) { if (t < st) red[t] += red[t + st]; __syncthreads(); }
    const float var = red[0] * (1.f / 1024.f);
    const float rinv = 1.0f / sqrtf(var + 1e-6f);
#pragma unroll
    for (int u = 0; u < 4; ++u) {
        int c = t + u * 256;
        out[(size_t)row * 1024 + c] = (v[u] - mu) * rinv * g[c] + bta[c];
    }
}

// ---------------------------------------------------------------------------
// Host orchestration
// ---------------------------------------------------------------------------
extern "C" void kernel_launch(void* const* d_in, const int* in_sizes, int n_in,
                              void* d_out, int out_size, void* d_ws, size_t ws_size,
                              hipStream_t stream)
{
    const float* X[3]   = { (const float*)d_in[0], (const float*)d_in[1], (const float*)d_in[2] };
    const int*   mask   = (const int*)d_in[3];
    const float* w_qs   = (const float*)d_in[4];
    const float* w_ks   = (const float*)d_in[5];
    const float* w_vs   = (const float*)d_in[6];
    const float* w_ds0  = (const float*)d_in[7];
    const float* w_ds1  = (const float*)d_in[8];
    const float* dc_w1  = (const float*)d_in[9];    // (2048, 2048)
    const float* dc_b1  = (const float*)d_in[10];
    const float* dc_w2  = (const float*)d_in[11];   // (2048, 2)
    const float* dc_b2  = (const float*)d_in[12];
    const float* w_fc   = (const float*)d_in[13];
    const float* ln1_g  = (const float*)d_in[14];
    const float* ln1_b  = (const float*)d_in[15];
    const float* ffn_w1 = (const float*)d_in[16];
    const float* ffn_b1 = (const float*)d_in[17];
    const float* ffn_w2 = (const float*)d_in[18];
    const float* ffn_b2 = (const float*)d_in[19];
    const float* ln2_g  = (const float*)d_in[20];
    const float* ln2_b  = (const float*)d_in[21];
    float* out = (float*)d_out;

    // workspace carve-up (floats)
    float* ws = (float*)d_ws;
    size_t off = 0;
    auto alloc = [&](size_t n) { float* p = ws + off; off += n; return p; };
    const size_t RM = (size_t)Rsz * DMsz;     // 256*1024
    float* Q   = alloc(3 * RM);
    float* Kb  = alloc(3 * RM);
    float* Vb  = alloc(3 * RM);
    float* d0  = alloc(3 * RM);
    float* d1  = alloc(3 * RM);
    float* a0  = alloc(3 * (size_t)Rsz * 2048);
    float* a1  = alloc(3 * (size_t)Rsz * 2048);
    float* dec = alloc(9 * (size_t)Bsz * Lsz * Lsz);
    float* S   = alloc((size_t)3 * Bsz * Hsz * Lsz * 192);
    float* O   = alloc(3 * RM);
    float* fc  = alloc(3 * RM);
    float* oln = alloc(3 * RM);
    float* h1  = alloc(3 * (size_t)Rsz * DIsz);
    float* f2  = alloc(3 * RM);
    (void)ws_size; (void)n_in; (void)in_sizes; (void)out_size;

    auto gemm = [&](const float* A, const float* Bm, const float* bias,
                    const float* resid, float* C, int M, int N, int K, int flags) {
        dim3 grid(N / 128, M / 64);
        gemm_wmma_f32<<<grid, dim3(256), 0, stream>>>(A, Bm, bias, resid, C, M, N, K, flags);
    };

    // ---- projections (all WMMA GEMMs) ----
    for (int i = 0; i < 3; ++i) {
        gemm(X[i], w_qs,  nullptr, nullptr, Q  + i * RM, Rsz, 1024, 1024, 0);
        gemm(X[i], w_ks,  nullptr, nullptr, Kb + i * RM, Rsz, 1024, 1024, 0);
        gemm(X[i], w_vs,  nullptr, nullptr, Vb + i * RM, Rsz, 1024, 1024, 0);
        gemm(X[i], w_ds0, nullptr, nullptr, d0 + i * RM, Rsz, 1024, 1024, 0);
        gemm(X[i], w_ds1, nullptr, nullptr, d1 + i * RM, Rsz, 1024, 1024, 0);
        // a0_i = d0_i @ dc_w1[D:]   (rows 1024..2047 of dc_w1, contiguous)
        gemm(d0 + i * RM, dc_w1 + (size_t)1024 * 2048, nullptr, nullptr,
             a0 + i * (size_t)Rsz * 2048, Rsz, 2048, 1024, 0);
        // a1_i = d1_i @ dc_w1[:D]
        gemm(d1 + i * RM, dc_w1, nullptr, nullptr,
             a1 + i * (size_t)Rsz * 2048, Rsz, 2048, 1024, 0);
    }

    // ---- decision gate (reduced analytically to a sign test) ----
    {
        int total_waves = 9 * Bsz * Lsz * Lsz;          // 147456
        dec_kernel<<<total_waves / 8, 256, 0, stream>>>(a0, a1, dc_b1, dc_w2, dc_b2, dec);
    }

    // ---- attention: scores+softmax, then A@V ----
    {
        int nblk = 3 * Bsz * Hsz * Lsz;                 // 12288
        attn_scores_kernel<<<nblk, 192, 0, stream>>>(Q, Kb, dec, mask, S);
        attn_av_kernel<<<nblk, 64, 0, stream>>>(S, Vb, O);
    }

    // ---- output projection + FFN + LayerNorms ----
    for (int i = 0; i < 3; ++i) {
        gemm(O + i * RM, w_fc, nullptr, X[i], fc + i * RM, Rsz, 1024, 1024, GEMM_RESID);
        layernorm_kernel<<<Rsz, 256, 0, stream>>>(fc + i * RM, nullptr, ln1_g, ln1_b, oln + i * RM);
        gemm(oln + i * RM, ffn_w1, ffn_b1, nullptr, h1 + i * (size_t)Rsz * DIsz,
             Rsz, DIsz, 1024, GEMM_BIAS | GEMM_RELU);
        gemm(h1 + i * (size_t)Rsz * DIsz, ffn_w2, ffn_b2, nullptr, f2 + i * RM,
             Rsz, 1024, DIsz, GEMM_BIAS);
        layernorm_kernel<<<Rsz, 256, 0, stream>>>(f2 + i * RM, oln + i * RM, ln2_g, ln2_b,
                                                  out + i * RM);
    }
}